// VectorQuantizer_13469017440757
// MI455X (gfx1250) — compile-verified
//
#include <hip/hip_runtime.h>
#include <hip/hip_bf16.h>

typedef __attribute__((ext_vector_type(2))) float v2f;
typedef __attribute__((ext_vector_type(8))) float v8f;

#define NUM_EMB   2048
#define EMB_DIM   64
#define N_ROWS    32768          // 32*32*32
#define QUANT_ELEMS (N_ROWS * EMB_DIM)

// ---------------------------------------------------------------------------
// Kernel 0: codebook squared norms  ||e||^2  per code
// ---------------------------------------------------------------------------
__global__ void vq_cbnorm_kernel(const float* __restrict__ cb,
                                 float* __restrict__ norms) {
    int k = blockIdx.x * blockDim.x + threadIdx.x;
    if (k < NUM_EMB) {
        const float* row = cb + k * EMB_DIM;
        float s = 0.0f;
#pragma unroll
        for (int d = 0; d < EMB_DIM; ++d) s += row[d] * row[d];
        norms[k] = s;
    }
}

// ---------------------------------------------------------------------------
// Kernel 1: zero the histogram
// ---------------------------------------------------------------------------
__global__ void vq_zero_counts_kernel(int* __restrict__ counts) {
    int k = blockIdx.x * blockDim.x + threadIdx.x;
    if (k < NUM_EMB) counts[k] = 0;
}

// ---------------------------------------------------------------------------
// Kernel 2: WMMA argmin.  Each wave owns 32 rows of x (two 16-row A tiles,
// two accumulators) and sweeps all 2048 codes in 16-code tiles with
// V_WMMA_F32_16X16X4_F32 (D=64 -> 16 chained k=4 steps per tile).
// Each B fragment feeds 2 WMMAs -> 2:1 compute:load in the hot loop.
// dist = ||e||^2 - 2 * (z . e)   (||z||^2 dropped; constant per row)
// ---------------------------------------------------------------------------
__global__ __launch_bounds__(256)
void vq_argmin_kernel(const float* __restrict__ X,
                      const float* __restrict__ CB,
                      const float* __restrict__ cbNorm,
                      int* __restrict__ outIdx) {
    const int lane = threadIdx.x & 31;
    const int wave = threadIdx.x >> 5;
    const int m    = lane & 15;      // row (A) / code-in-tile (B) / N col (D)
    const int kh   = lane >> 4;      // which k-pair half
    const int rowBase = blockIdx.x * 256 + wave * 32;

    // ---- load A fragments for two 16-row tiles, 16 steps of K=4 each.
    // A layout (16x4 f32): lanes 0-15 hold k={4s,4s+1}, lanes 16-31 k={4s+2,4s+3}
    v2f a0[16], a1[16];
    {
        const float2* xr0 =
            reinterpret_cast<const float2*>(X + (size_t)(rowBase + m) * EMB_DIM);
        const float2* xr1 =
            reinterpret_cast<const float2*>(X + (size_t)(rowBase + 16 + m) * EMB_DIM);
#pragma unroll
        for (int s = 0; s < 16; ++s) {
            float2 t0 = xr0[s * 2 + kh];
            float2 t1 = xr1[s * 2 + kh];
            a0[s].x = t0.x; a0[s].y = t0.y;
            a1[s].x = t1.x; a1[s].y = t1.y;
        }
    }

    float best0[8], best1[8];
    int   bestIdx0[8], bestIdx1[8];
#pragma unroll
    for (int j = 0; j < 8; ++j) {
        best0[j] = 3.402823e38f; bestIdx0[j] = 0;
        best1[j] = 3.402823e38f; bestIdx1[j] = 0;
    }

    // ---- sweep code tiles: 2048 codes / 16 per tile = 128 iterations
    for (int ct = 0; ct < NUM_EMB / 16; ++ct) {
        const int code = ct * 16 + m;
        // B = codebook^T (4x16 per step); same float2 pattern on codebook rows.
        const float2* crow =
            reinterpret_cast<const float2*>(CB + (size_t)code * EMB_DIM);
        // pull the codebook row 8 codes ahead toward the WGP while this tile computes
        if (ct + 1 < NUM_EMB / 16)
            __builtin_prefetch(crow + 16 * (EMB_DIM / 4), 0, 3);

        v2f b[16];
#pragma unroll
        for (int s = 0; s < 16; ++s) {
            float2 t = crow[s * 2 + kh];
            b[s].x = t.x; b[s].y = t.y;
        }

        v8f acc0 = {}, acc1 = {};
#pragma unroll
        for (int s = 0; s < 16; ++s) {
            acc0 = __builtin_amdgcn_wmma_f32_16x16x4_f32(
                false, a0[s], false, b[s], (short)0, acc0, false, false);
            acc1 = __builtin_amdgcn_wmma_f32_16x16x4_f32(
                false, a1[s], false, b[s], (short)0, acc1, false, false);
        }

        const float cn = cbNorm[code];
        // D layout: VGPR j -> lanes 0-15: M=j, lanes 16-31: M=j+8; N = lane&15
#pragma unroll
        for (int j = 0; j < 8; ++j) {
            float d0 = __builtin_fmaf(-2.0f, acc0[j], cn);
            float d1 = __builtin_fmaf(-2.0f, acc1[j], cn);
            if (d0 < best0[j]) { best0[j] = d0; bestIdx0[j] = code; }
            if (d1 < best1[j]) { best1[j] = d1; bestIdx1[j] = code; }
        }
    }

    // ---- min+index reduction across the 16 lanes sharing a row
    // (xor masks 1,2,4,8 stay inside each 16-lane half of the wave)
#pragma unroll
    for (int off = 1; off < 16; off <<= 1) {
#pragma unroll
        for (int j = 0; j < 8; ++j) {
            float ov0 = __shfl_xor(best0[j],    off, 32);
            int   oi0 = __shfl_xor(bestIdx0[j], off, 32);
            if (ov0 < best0[j] || (ov0 == best0[j] && oi0 < bestIdx0[j])) {
                best0[j] = ov0; bestIdx0[j] = oi0;
            }
            float ov1 = __shfl_xor(best1[j],    off, 32);
            int   oi1 = __shfl_xor(bestIdx1[j], off, 32);
            if (ov1 < best1[j] || (ov1 == best1[j] && oi1 < bestIdx1[j])) {
                best1[j] = ov1; bestIdx1[j] = oi1;
            }
        }
    }

    if (m == 0) {
        // kh==0 lanes hold rows j (0..7); kh==1 lanes hold rows j+8
#pragma unroll
        for (int j = 0; j < 8; ++j) {
            outIdx[rowBase      + kh * 8 + j] = bestIdx0[j];
            outIdx[rowBase + 16 + kh * 8 + j] = bestIdx1[j];
        }
    }
}

// ---------------------------------------------------------------------------
// Kernel 3: gather quantized rows, per-block SSE partials (deterministic),
// histogram, and indices-as-float output.
// One thread per (n, d) element.
// ---------------------------------------------------------------------------
__global__ __launch_bounds__(256)
void vq_gather_kernel(const float* __restrict__ X,
                      const float* __restrict__ CB,
                      const int*  __restrict__ idx,
                      float* __restrict__ out,      // full d_out
                      float* __restrict__ partial,
                      int*   __restrict__ counts) {
    const int gid  = blockIdx.x * 256 + threadIdx.x;
    const int n    = gid >> 6;
    const int d    = gid & 63;
    const int lane = threadIdx.x & 31;
    const int wave = threadIdx.x >> 5;

    const int code = idx[n];
    const float q  = CB[(size_t)code * EMB_DIM + d];
    out[gid] = q;                                   // quantized_st == quantized
    const float diff = q - X[gid];
    float e = diff * diff;

    if (d == 0) {
        atomicAdd(&counts[code], 1);                // integer: deterministic
        out[QUANT_ELEMS + 3 + n] = (float)code;     // indices as float
    }

    // deterministic block reduction of e
#pragma unroll
    for (int off = 16; off > 0; off >>= 1) e += __shfl_xor(e, off, 32);
    __shared__ float red[8];
    if (lane == 0) red[wave] = e;
    __syncthreads();
    if (threadIdx.x == 0) {
        float s = 0.0f;
#pragma unroll
        for (int w = 0; w < 8; ++w) s += red[w];
        partial[blockIdx.x] = s;
    }
}

// ---------------------------------------------------------------------------
// Kernel 4: finalize losses + perplexity (single block, fixed-order sums)
// ---------------------------------------------------------------------------
__global__ __launch_bounds__(256)
void vq_finalize_kernel(const float* __restrict__ partial, int numPartial,
                        const int* __restrict__ counts,
                        float* __restrict__ out) {
    const int tid = threadIdx.x;
    __shared__ float red[256];

    // --- SSE -> mean squared error
    float s = 0.0f;
    for (int i = tid; i < numPartial; i += 256) s += partial[i];
    red[tid] = s;
    __syncthreads();
#pragma unroll
    for (int off = 128; off > 0; off >>= 1) {
        if (tid < off) red[tid] += red[tid + off];
        __syncthreads();
    }
    float loss = red[0] * (1.0f / (float)QUANT_ELEMS);
    __syncthreads();

    // --- entropy -> perplexity
    float h = 0.0f;
    for (int k = tid; k < NUM_EMB; k += 256) {
        float p = (float)counts[k] * (1.0f / (float)N_ROWS);
        h += p * logf(p + 1e-10f);
    }
    red[tid] = h;
    __syncthreads();
#pragma unroll
    for (int off = 128; off > 0; off >>= 1) {
        if (tid < off) red[tid] += red[tid + off];
        __syncthreads();
    }

    if (tid == 0) {
        out[QUANT_ELEMS + 0] = loss;            // vq_loss
        out[QUANT_ELEMS + 1] = loss;            // commit_loss (identical)
        out[QUANT_ELEMS + 2] = expf(-red[0]);   // perplexity
    }
}

// ---------------------------------------------------------------------------
extern "C" void kernel_launch(void* const* d_in, const int* in_sizes, int n_in,
                              void* d_out, int out_size, void* d_ws, size_t ws_size,
                              hipStream_t stream) {
    const float* X  = (const float*)d_in[0];   // [32,32,32,64]
    const float* CB = (const float*)d_in[1];   // [2048,64]
    float* out = (float*)d_out;

    // workspace layout
    char* ws = (char*)d_ws;
    float* cbNorm  = (float*)(ws);                     //  2048 f32 ->   8 KB
    int*   indices = (int*)  (ws + 8192);              // 32768 i32 -> 128 KB
    int*   counts  = (int*)  (ws + 8192 + 131072);     //  2048 i32 ->   8 KB
    float* partial = (float*)(ws + 8192 + 131072 + 8192); // 8192 f32 -> 32 KB

    const int nPartial = QUANT_ELEMS / 256;            // 8192 blocks

    vq_cbnorm_kernel<<<(NUM_EMB + 255) / 256, 256, 0, stream>>>(CB, cbNorm);
    vq_zero_counts_kernel<<<(NUM_EMB + 255) / 256, 256, 0, stream>>>(counts);
    vq_argmin_kernel<<<N_ROWS / 256, 256, 0, stream>>>(X, CB, cbNorm, indices);
    vq_gather_kernel<<<nPartial, 256, 0, stream>>>(X, CB, indices, out, partial, counts);
    vq_finalize_kernel<<<1, 256, 0, stream>>>(partial, nPartial, counts, out);
}